// MUCMA_16535624089791
// MI455X (gfx1250) — compile-verified
//
#include <hip/hip_runtime.h>

// MUCMA serial scan, single wave32, CDNA5 (gfx1250).
// - async global->LDS ring buffer (DEPTH=8) hides HBM latency of the per-step frame
// - v / s reductions: 5-stage wave32 butterflies (ds_bpermute)
// - r EWMA (beta*r + (1-beta)*v (x) conj(z)) done by ONE v_wmma_f32_16x16x4_f32 per step
// - bias-correction reciprocal via v_rcp_f32 (serial critical path, full div not needed)
// Output sig[n][i][re/im] == v_n (pre-update weights), matching the reference.

#define NSYM   131072
#define TAPS   19
#define DIMS   2
#define DELTA  6
#define FPS    (TAPS * DIMS)      // 38 floats per component row
#define LRc    (2.44140625e-4f)   // 2^-12
#define BETAc  (0.999f)
#define R2c    (1.32f)

#define DEPTH       8             // LDS ring depth (iterations in flight)
#define SLOT_FLOATS 80            // 38 re + 38 im, padded to 80 floats (320 B)

typedef __attribute__((ext_vector_type(2))) float v2f;
typedef __attribute__((ext_vector_type(8))) float v8f;
typedef __attribute__((address_space(3))) float lds_f;   // 32-bit LDS offset pointer

// ---- CDNA5 async global->LDS copy (ASYNCcnt-tracked), inline asm ----
// The AS(3) pointer operand keeps sm's provenance alive (no poison-folding of LDS
// reads) AND gives the instruction its LDS byte-offset VGPR directly.
__device__ __forceinline__ void async_copy_b64(lds_f* lp, const float* gp) {
    asm volatile("global_load_async_to_lds_b64 %0, %1, off"
                 :: "v"(lp), "v"(gp)
                 : "memory");
}
__device__ __forceinline__ void wait_asynccnt_le14() {
    asm volatile("s_wait_asynccnt 0xe" ::: "memory");
}
__device__ __forceinline__ void wait_dscnt_0() {
    asm volatile("s_wait_dscnt 0x0" ::: "memory");
}

__global__ void __launch_bounds__(32)
mucma_scan_kernel(const float* __restrict__ fr_re,
                  const float* __restrict__ fr_im,
                  const float* __restrict__ w0_re,
                  const float* __restrict__ w0_im,
                  float* __restrict__ out)
{
    __shared__ float sm[DEPTH * SLOT_FLOATS];

    // Belt-and-braces: force sm to escape into an asm that may write memory, so the
    // optimizer can never conclude "sm has no stores" and fold its loads to poison.
    {
        float* esc = &sm[0];
        asm volatile("" : "+v"(esc) :: "memory");
    }
    lds_f* lbase = (lds_f*)&sm[0];

    const int  lane   = threadIdx.x;      // 0..31, one wave
    const int  n16    = lane & 15;
    const bool lohalf = (lane < 16);
    const int  lcol   = n16 & 1;          // output dim for this lane's (l,d) column

    // Two (tap,dim) flat slots per lane: f0 = lane (<38 always), f1 = lane+32 (lanes 0..5)
    const int  f0   = lane;
    const int  f1   = lane + 32;
    const bool has1 = (f1 < FPS);

    // --- load initial weights for my slots: w[k][slot], k = output dim ---
    float wr[DIMS][2], wi[DIMS][2];
    {
        const int t0 = f0 >> 1, j0 = f0 & 1;
        const int t1 = f1 >> 1, j1 = f1 & 1;
        #pragma unroll
        for (int k = 0; k < DIMS; ++k) {
            wr[k][0] = w0_re[(k * DIMS + j0) * TAPS + t0];
            wi[k][0] = w0_im[(k * DIMS + j0) * TAPS + t0];
            wr[k][1] = has1 ? w0_re[(k * DIMS + j1) * TAPS + t1] : 0.f;
            wi[k][1] = has1 ? w0_im[(k * DIMS + j1) * TAPS + t1] : 0.f;
        }
    }

    // per-lane delay-line value: lane n16 = d*2+lcol holds z[d][lcol]; lanes<16 real, >=16 imag
    float my_z = 0.f;
    // WMMA accumulator: realified r[k,(lcol,d)] (rows 0..3 of lanes 0..11), rest zero
    v8f racc = {};
    float bp = BETAc;

    // --- prologue: stage first DEPTH frames into the LDS ring (2 async b64 per frame) ---
    if (lane < TAPS) {
        #pragma unroll
        for (int i = 0; i < DEPTH; ++i) {
            lds_f* slotp = lbase + i * SLOT_FLOATS;
            async_copy_b64(slotp + lane * 2,
                           fr_re + (size_t)i * FPS + lane * 2);
            async_copy_b64(slotp + FPS + lane * 2,
                           fr_im + (size_t)i * FPS + lane * 2);
        }
    }

    const float s1b = 1.0f - BETAc;

    for (int n = 0; n < NSYM; ++n) {
        // wait until the oldest staged frame (iteration n) has landed in LDS
        wait_asynccnt_le14();
        const int slot = n & (DEPTH - 1);
        const float* srow = &sm[slot * SLOT_FLOATS];

        float ur0 = srow[f0];
        float ui0 = srow[FPS + f0];
        float ur1 = 0.f, ui1 = 0.f;
        if (has1) { ur1 = srow[f1]; ui1 = srow[FPS + f1]; }

        // make sure our ds reads retired before re-filling this slot
        wait_dscnt_0();
        const int np = n + DEPTH;
        if (np < NSYM && lane < TAPS) {
            lds_f* slotp = lbase + slot * SLOT_FLOATS;
            async_copy_b64(slotp + lane * 2,
                           fr_re + (size_t)np * FPS + lane * 2);
            async_copy_b64(slotp + FPS + lane * 2,
                           fr_im + (size_t)np * FPS + lane * 2);
        }

        // ---- v[k] = sum_f w[k,f] * u[f] : per-lane partials + butterfly reduce ----
        float pv0 = wr[0][0]*ur0 - wi[0][0]*ui0 + wr[0][1]*ur1 - wi[0][1]*ui1;
        float pv1 = wr[0][0]*ui0 + wi[0][0]*ur0 + wr[0][1]*ui1 + wi[0][1]*ur1;
        float pv2 = wr[1][0]*ur0 - wi[1][0]*ui0 + wr[1][1]*ur1 - wi[1][1]*ui1;
        float pv3 = wr[1][0]*ui0 + wi[1][0]*ur0 + wr[1][1]*ui1 + wi[1][1]*ur1;
        #pragma unroll
        for (int m = 16; m >= 1; m >>= 1) {
            pv0 += __shfl_xor(pv0, m, 32);
            pv1 += __shfl_xor(pv1, m, 32);
            pv2 += __shfl_xor(pv2, m, 32);
            pv3 += __shfl_xor(pv3, m, 32);
        }
        const float v0r = pv0, v0i = pv1, v1r = pv2, v1i = pv3;

        // ---- emit output: sig[n] = v (pre-update weights) ----
        if (lane == 0) {
            *(float4*)(out + (size_t)n * 4) = make_float4(v0r, v0i, v1r, v1i);
        }

        // ---- delay-line pipeline: z = concat(v, z[:-1]) as lane-to-lane shift by 2 ----
        const float shifted = __shfl_up(my_z, 2, 32);
        const float fresh   = lohalf ? (lcol ? v1r : v0r) : (lcol ? v1i : v0i);
        my_z = (n16 < 2) ? fresh : ((n16 < 12) ? shifted : 0.f);
        const float z_r = my_z;                        // valid in lanes < 16
        const float z_i = __shfl_xor(my_z, 16, 32);    // partner's imag (lanes < 16)

        // ---- r <- beta*r + (1-beta) * v (x) conj(z)  via ONE f32 WMMA ----
        // A (16x4): rows 0..3 = realified (1-beta)*v : [vr0 vi0 | vi0 -vr0 | vr1 vi1 | vi1 -vr1]
        float a0 = 0.f, a1 = 0.f;
        if (lohalf) {                                   // lanes<16 carry K=0,1; K=2,3 pad zero
            if      (n16 == 0) { a0 =  v0r * s1b; a1 =  v0i * s1b; }
            else if (n16 == 1) { a0 =  v0i * s1b; a1 = -v0r * s1b; }
            else if (n16 == 2) { a0 =  v1r * s1b; a1 =  v1i * s1b; }
            else if (n16 == 3) { a0 =  v1i * s1b; a1 = -v1r * s1b; }
        }
        v2f A; A[0] = a0; A[1] = a1;
        // B (4x16): rows 0/1 = zr/zi (lanes<16 carry K=0,1); K=2,3 pad zero (lanes>=16)
        v2f B; B[0] = lohalf ? z_r : 0.f; B[1] = lohalf ? z_i : 0.f;
        #pragma unroll
        for (int q = 0; q < 8; ++q) racc[q] *= BETAc;   // beta decay of accumulator
        racc = __builtin_amdgcn_wmma_f32_16x16x4_f32(
                   false, A, false, B, (short)0, racc, false, false);

        // ---- s[k] = sum_{l!=k,d} rhat[k,l,d] * conj(z[d,l]) ----
        // lane n16=d*2+lcol contributes to k=1-lcol; rr/ri from D rows (racc[0..3])
        const float rr = lcol ? racc[0] : racc[2];
        const float ri = lcol ? racc[1] : racc[3];
        const float psre = rr * z_r + ri * z_i;        // zero in lanes>=16 / n16>=12 (racc=0)
        const float psim = ri * z_r - rr * z_i;
        float s0r = lcol ? psre : 0.f;
        float s0i = lcol ? psim : 0.f;
        float s1r = lcol ? 0.f : psre;
        float s1i = lcol ? 0.f : psim;
        #pragma unroll
        for (int m = 16; m >= 1; m >>= 1) {
            s0r += __shfl_xor(s0r, m, 32);
            s0i += __shfl_xor(s0i, m, 32);
            s1r += __shfl_xor(s1r, m, 32);
            s1i += __shfl_xor(s1i, m, 32);
        }
        // bias correction (rhat = r/(1-bp)): v_rcp_f32 is plenty for this factor and
        // keeps ~10 dependent div-fixup ops off the serial critical path
        const float inv = __builtin_amdgcn_rcpf(1.0f - bp);

        // ---- CMA error + combined update coefficient c[k] = 2*LR*(e[k]*v[k] - s[k]) ----
        const float e0 = R2c - (v0r * v0r + v0i * v0i);
        const float e1 = R2c - (v1r * v1r + v1i * v1i);
        const float g  = 2.0f * LRc;
        const float c0r = g * (e0 * v0r - s0r * inv);
        const float c0i = g * (e0 * v0i - s0i * inv);
        const float c1r = g * (e1 * v1r - s1r * inv);
        const float c1i = g * (e1 * v1i - s1i * inv);

        // ---- rank-1 weight update: w[k,f] += c[k] * conj(u[f]) ----
        wr[0][0] += c0r * ur0 + c0i * ui0;  wi[0][0] += c0i * ur0 - c0r * ui0;
        wr[0][1] += c0r * ur1 + c0i * ui1;  wi[0][1] += c0i * ur1 - c0r * ui1;
        wr[1][0] += c1r * ur0 + c1i * ui0;  wi[1][0] += c1i * ur0 - c1r * ui0;
        wr[1][1] += c1r * ur1 + c1i * ui1;  wi[1][1] += c1i * ur1 - c1r * ui1;

        bp *= BETAc;
    }
}

extern "C" void kernel_launch(void* const* d_in, const int* in_sizes, int n_in,
                              void* d_out, int out_size, void* d_ws, size_t ws_size,
                              hipStream_t stream) {
    (void)in_sizes; (void)n_in; (void)out_size; (void)d_ws; (void)ws_size;
    const float* fr_re = (const float*)d_in[0];
    const float* fr_im = (const float*)d_in[1];
    const float* w0_re = (const float*)d_in[2];
    const float* w0_im = (const float*)d_in[3];
    float* out = (float*)d_out;
    hipLaunchKernelGGL(mucma_scan_kernel, dim3(1), dim3(32), 0, stream,
                       fr_re, fr_im, w0_re, w0_im, out);
}